// DifferentiablePersistence_57964878627367
// MI455X (gfx1250) — compile-verified
//
#include <hip/hip_runtime.h>
#include <stdint.h>

#define BB 8
#define CC 64
#define HH 256
#define WW 256
#define LL 16
#define NSTEPS 5
#define TEMPR 10.0f
#define TILE 22          // output tile (22x22)
#define HALO 5
#define RG 32            // region = TILE + 2*HALO = one column per lane
#define RC (RG*RG)       // 1024
#define NT 12            // ceil(256/22) tiles per axis
#define CTR (TILE*TILE)  // 484 center pixels

typedef unsigned int tdm_u32x4 __attribute__((ext_vector_type(4)));
typedef int tdm_i32x8 __attribute__((ext_vector_type(8)));
typedef int tdm_i32x4 __attribute__((ext_vector_type(4)));

__device__ __forceinline__ float sigf(float x) { return 1.0f / (1.0f + __expf(-x)); }

// ---------------------------------------------------------------------------
// Kernel 1: per-batch min / range of filtration
// ---------------------------------------------------------------------------
__global__ void dp_minmax_kernel(const float* __restrict__ filt, float* __restrict__ ws) {
    __shared__ float smn[256], smx[256];
    const int b = blockIdx.x, t = threadIdx.x;
    const float* p = filt + (size_t)b * HH * WW;
    float mn = 3.4e38f, mx = -3.4e38f;
    for (int i = t; i < HH * WW; i += 256) {
        float v = p[i];
        mn = fminf(mn, v);
        mx = fmaxf(mx, v);
    }
    smn[t] = mn; smx[t] = mx;
    __syncthreads();
    for (int s = 128; s > 0; s >>= 1) {
        if (t < s) {
            smn[t] = fminf(smn[t], smn[t + s]);
            smx[t] = fmaxf(smx[t], smx[t + s]);
        }
        __syncthreads();
    }
    if (t == 0) {
        ws[b * 2 + 0] = smn[0];
        ws[b * 2 + 1] = fmaxf(smx[0] - smn[0], 1e-6f);
    }
}

// ---------------------------------------------------------------------------
// Kernel 2: main fused persistence/diffusion kernel
// grid = (NT, NT, B), block = 256 threads (8 waves); lane = region column
// ---------------------------------------------------------------------------
__global__ void __launch_bounds__(256)
dp_persist_kernel(const float* __restrict__ filt, const float* __restrict__ feat,
                  const float* __restrict__ lvloff, const float* __restrict__ ws,
                  float* __restrict__ out) {
    __shared__ float s_filt[RC + 8];          // filtration region (1e30 outside image)
    __shared__ float2 s_mrd[LL * RC + 4];     // per level: {0.5*mask, 1/max(summask,1)}
    __shared__ float s_var[8 * CTR];          // per-wave variance partial sums
    __shared__ float s_thr[LL];
    __shared__ float s_lev[LL];

    const int tx = blockIdx.x, ty = blockIdx.y, b = blockIdx.z;
    const int t = threadIdx.x;
    const int wave = t >> 5, lane = t & 31;
    const int gx0 = tx * TILE - HALO, gy0 = ty * TILE - HALO;

    // --- thresholds / levels -------------------------------------------------
    if (t < LL) {
        float lv = sigf(lvloff[t]);            // monotonic input -> already sorted
        s_lev[t] = lv;
        s_thr[t] = ws[b * 2 + 0] + lv * ws[b * 2 + 1];
    }

    // --- load filtration region: TDM for interior tiles, async-LDS otherwise
    const float* fb = filt + (size_t)b * HH * WW;
    const bool interior = (gx0 >= 0) && (gy0 >= 0) && (gx0 + RG <= WW) && (gy0 + RG <= HH);
    if (interior) {
        if (wave == 0) {
            uint64_t ga = (uint64_t)(uintptr_t)(fb + (size_t)gy0 * WW + gx0);
            uint32_t lds = (uint32_t)(uintptr_t)(&s_filt[0]);
            tdm_u32x4 g0;
            g0[0] = 1u;                                            // count=1, user mode
            g0[1] = lds;                                           // lds_addr
            g0[2] = (uint32_t)ga;                                  // global_addr[31:0]
            g0[3] = (uint32_t)((ga >> 32) & 0x01FFFFFFu) | 0x80000000u; // [56:32] | type=2
            tdm_i32x8 g1;
            g1[0] = (int)(2u << 16);                               // data_size = 4 bytes
            g1[1] = (int)((uint32_t)RG << 16);                     // tensor_dim0 = 32
            g1[2] = (int)((uint32_t)RG << 16);                     // tensor_dim1 = 32
            g1[3] = (int)((uint32_t)RG << 16);                     // tile_dim0 = 32
            g1[4] = RG;                                            // tile_dim1 = 32
            g1[5] = WW;                                            // tensor_dim0_stride
            g1[6] = 0;
            g1[7] = 0;
            tdm_i32x4 gz = {0, 0, 0, 0};
#if __clang_major__ >= 23
            tdm_i32x8 gz8 = {0, 0, 0, 0, 0, 0, 0, 0};
            __builtin_amdgcn_tensor_load_to_lds(g0, g1, gz, gz, gz8, 0);
#else
            __builtin_amdgcn_tensor_load_to_lds(g0, g1, gz, gz, 0);
#endif
            __builtin_amdgcn_s_wait_tensorcnt(0);
        }
    } else {
        for (int k = t; k < RC; k += 256) {
            int y = k >> 5, x = k & 31;
            int gy = gy0 + y, gx = gx0 + x;
            bool in = (gy >= 0) && (gy < HH) && (gx >= 0) && (gx < WW);
            if (in) {
                uint32_t ldsoff = (uint32_t)(uintptr_t)(&s_filt[k]);
                uint64_t ga = (uint64_t)(uintptr_t)(fb + (size_t)gy * WW + gx);
                asm volatile("global_load_async_to_lds_b32 %0, %1, off"
                             :: "v"(ldsoff), "v"(ga) : "memory");
            } else {
                s_filt[k] = 1e30f;                 // sigmoid -> exact 0 outside image
            }
        }
        asm volatile("s_wait_asynccnt 0" ::: "memory");
    }
    __syncthreads();

    // --- masks (pre-halved) + rd default ------------------------------------
    for (int i = t; i < LL * RC; i += 256) {
        int l = i >> 10, k = i & (RC - 1);
        float m = sigf(TEMPR * (s_thr[l] - s_filt[k]));
        s_mrd[i] = make_float2(0.5f * m, 1.0f);
    }
    __syncthreads();

    // --- rd interior: 1 / max(sum of 4 neighbor masks, 1) (no div/mod) ------
    for (int i = t; i < LL * RC; i += 256) {
        int k = i & (RC - 1);
        int y = k >> 5, x = k & 31;
        if (y >= 1 && y < RG - 1 && x >= 1 && x < RG - 1) {
            float sm = 2.0f * (s_mrd[i - 1].x + s_mrd[i + 1].x +
                               s_mrd[i - RG].x + s_mrd[i + RG].x);
            s_mrd[i].y = 1.0f / fmaxf(sm, 1.0f);
        }
    }

    // --- birth map: thread t owns center pixels t and t+256 -----------------
    float birth0 = 0.0f, birth1 = 0.0f;
    {
        int p0 = t, p1 = t + 256;
        int k0 = (p0 / TILE + HALO) * RG + (p0 % TILE + HALO);
        int k1 = (p1 / TILE + HALO) * RG + (p1 % TILE + HALO);
        float pm0 = 0.0f, pm1 = 0.0f;
        for (int l = 0; l < LL; ++l) {
            float m0 = 2.0f * s_mrd[l * RC + k0].x;
            birth0 = fmaf(fmaxf(m0 - pm0, 0.0f), s_lev[l], birth0);
            pm0 = m0;
            if (p1 < CTR) {
                float m1 = 2.0f * s_mrd[l * RC + k1].x;
                birth1 = fmaf(fmaxf(m1 - pm1, 0.0f), s_lev[l], birth1);
                pm1 = m1;
            }
        }
    }
    __syncthreads();

    // --- main loop: lane = column x of the 32-wide region -------------------
    const int gx = gx0 + lane;
    const bool gxv = (gx >= 0) && (gx < WW);
    const bool center = (lane >= HALO) && (lane < HALO + TILE);
    const int px = lane - HALO;

    // per-lane prefetch row for the next channel's tile (one row per lane)
    int pfy = gy0 + lane; pfy = (pfy < 0) ? 0 : ((pfy > HH - 1) ? HH - 1 : pfy);
    int pfx = (gx0 < 0) ? 0 : ((gx0 > WW - 1) ? WW - 1 : gx0);

    float varsum[TILE];
#pragma unroll
    for (int py = 0; py < TILE; ++py) varsum[py] = 0.0f;

    for (int ci = 0; ci < CC / 8; ++ci) {
        const int c = ci * 8 + wave;
        const float* fc = feat + (((size_t)b * CC + c) * HH) * WW;

        float feat2[RG];                           // 2 * feature column (0 outside image)
#pragma unroll
        for (int y = 0; y < RG; ++y) {
            int gy = gy0 + y;
            feat2[y] = (gxv && gy >= 0 && gy < HH) ? 2.0f * fc[(size_t)gy * WW + gx] : 0.0f;
        }
        if (ci + 1 < CC / 8) {                     // CDNA5 global_prefetch: each lane
            __builtin_prefetch(fc + (size_t)8 * HH * WW + (size_t)pfy * WW + pfx, 0, 0);
        }                                          //   fetches one row of next tile

        float mean[TILE], m2[TILE];
#pragma unroll
        for (int py = 0; py < TILE; ++py) { mean[py] = 0.0f; m2[py] = 0.0f; }

        for (int l = 0; l < LL; ++l) {
            const float2* mrd = &s_mrd[l * RC];
            float mhr[RG], rdr[RG], f[RG];
#pragma unroll
            for (int y = 0; y < RG; ++y) {         // one ds_load_b64 per row
                float2 v = mrd[y * RG + lane];
                mhr[y] = v.x;
                rdr[y] = v.y;
                f[y] = feat2[y] * mhr[y];          // field0 = feat * mask
            }
            // 5 diffusion steps; exact shrinking valid region; field in registers.
            // vertical neighbors in-register, horizontal via cross-lane shuffle.
#pragma unroll
            for (int s = 0; s < NSTEPS; ++s) {
                float prev = f[s];
#pragma unroll
                for (int y = s + 1; y <= 30 - s; ++y) {
                    float cur = f[y];
                    float nxt = f[y + 1];
                    float lft = __shfl_up(cur, 1);
                    float rgt = __shfl_down(cur, 1);
                    float sum = (prev + nxt) + (lft + rgt);
                    f[y] = mhr[y] * fmaf(sum, rdr[y], cur); // mask*(0.5f + 0.5conv/denom)
                    prev = cur;
                }
            }
            // Welford over levels
            float rcnt = 1.0f / (float)(l + 1);
#pragma unroll
            for (int py = 0; py < TILE; ++py) {
                float comp = f[py + HALO];
                float d = comp - mean[py];
                mean[py] = fmaf(d, rcnt, mean[py]);
                m2[py] = fmaf(d, comp - mean[py], m2[py]);
                if (l == LL - 1 && center) {        // last_comp output
                    int row = ty * TILE + py, col = tx * TILE + px;
                    if (row < HH && col < WW)
                        out[(size_t)3 * BB * HH * WW +
                            (((size_t)b * CC + c) * HH + (size_t)row) * WW + col] = comp;
                }
            }
        }
#pragma unroll
        for (int py = 0; py < TILE; ++py)
            varsum[py] = fmaf(m2[py], 1.0f / (float)(LL - 1), varsum[py]);
    }

    // deterministic cross-wave variance reduction via disjoint LDS slices
    if (center) {
#pragma unroll
        for (int py = 0; py < TILE; ++py)
            s_var[wave * CTR + py * TILE + px] = varsum[py];
    }
    __syncthreads();

    // --- finalize center pixels (2 per thread) ------------------------------
    for (int pi = 0; pi < 2; ++pi) {
        int p = t + pi * 256;
        if (p >= CTR) break;
        float vs = 0.0f;
#pragma unroll
        for (int w = 0; w < 8; ++w) vs += s_var[w * CTR + p];
        float vmean = vs * (1.0f / (float)CC);
        float stab = 1.0f / (1.0f + vmean);
        float birth = (pi == 0) ? birth0 : birth1;
        float pers = (1.0f - birth) * stab;
        int row = ty * TILE + p / TILE, col = tx * TILE + p % TILE;
        if (row < HH && col < WW) {
            size_t pix = ((size_t)b * HH + (size_t)row) * WW + col;
            out[pix] = pers;
            out[(size_t)BB * HH * WW + pix] = birth;
            out[(size_t)2 * BB * HH * WW + pix] = stab;
        }
    }
}

// ---------------------------------------------------------------------------
extern "C" void kernel_launch(void* const* d_in, const int* in_sizes, int n_in,
                              void* d_out, int out_size, void* d_ws, size_t ws_size,
                              hipStream_t stream) {
    const float* filtration = (const float*)d_in[0];
    const float* features = (const float*)d_in[1];
    const float* level_offsets = (const float*)d_in[2];
    float* ws = (float*)d_ws;
    float* o = (float*)d_out;
    (void)in_sizes; (void)n_in; (void)out_size; (void)ws_size;

    dp_minmax_kernel<<<dim3(BB), dim3(256), 0, stream>>>(filtration, ws);
    dp_persist_kernel<<<dim3(NT, NT, BB), dim3(256), 0, stream>>>(
        filtration, features, level_offsets, ws, o);
}